// PGLoss_39900246180103
// MI455X (gfx1250) — compile-verified
//
#include <hip/hip_runtime.h>

// PGLoss: loss = -sum_{b,s} logprobs[b,s,label[b,s]] * reward[b,s]
//
// logprobs: [4,2048,32000] f32 (~1.05 GB) -- only 8192 gathered elements needed.
// Latency-bound gather kernel: one workgroup (32 wave32s), 8 tokens/thread,
// deterministic reduction (shfl_xor -> LDS -> V_WMMA_F32_16X16X4_F32 -> scalar).

typedef __attribute__((ext_vector_type(2))) float v2f;
typedef __attribute__((ext_vector_type(8))) float v8f;

__global__ __launch_bounds__(1024) void pg_loss_kernel(
    const float* __restrict__ logprobs,
    const int*   __restrict__ label,
    const float* __restrict__ reward,
    float*       __restrict__ out,
    int n, int vocab)
{
    const int tid  = threadIdx.x;      // 0..1023
    const int lane = tid & 31;
    const int wave = tid >> 5;

    // ---- Per-thread gather + weighted accumulate (fixed order => deterministic)
    float acc = 0.0f;
#pragma unroll
    for (int k = 0; k < 8; ++k) {
        int i = tid + k * 1024;
        if (i < n) {
            int   lbl = label[i];                               // coalesced
            float rw  = reward[i];                              // coalesced
            float lp  = logprobs[(size_t)i * (size_t)vocab + (size_t)lbl]; // scattered
            acc = fmaf(lp, rw, acc);
        }
    }

    // ---- Wave32 butterfly reduction (lane-permute ops, fixed order)
#pragma unroll
    for (int off = 16; off > 0; off >>= 1)
        acc += __shfl_xor(acc, off, 32);

    // ---- Stage one partial per wave in LDS
    __shared__ float wsum[32];
    if (lane == 0) wsum[wave] = acc;
    __syncthreads();

    // ---- Final 32 -> 1 reduction in wave 0 via V_WMMA_F32_16X16X4_F32.
    // A (16x4 f32 layout): lanes 0-15 hold {A[m][0], A[m][1]} with m=lane,
    //                      lanes 16-31 hold {A[m][2], A[m][3]} with m=lane-16.
    // Put partial[lane] in K-slot 0/2, zero in K-slot 1/3; B = ones(4x16).
    // => D[m][n] = partial[m] + partial[m+16] for every n.
    if (wave == 0) {
        float p = wsum[lane];
        v2f a;  a[0] = p;    a[1] = 0.0f;
        v2f b;  b[0] = 1.0f; b[1] = 1.0f;   // all-ones 4x16 B
        v8f c = {};
        // 8 args: (neg_a, A, neg_b, B, c_mod, C, reuse_a, reuse_b)
        c = __builtin_amdgcn_wmma_f32_16x16x4_f32(
                /*neg_a=*/false, a, /*neg_b=*/false, b,
                /*c_mod=*/(short)0, c, /*reuse_a=*/false, /*reuse_b=*/false);
        // Lane 0 holds D[m=j][n=0] in c[j] (rows 0..7); lane 16 holds rows 8..15.
        float s = c[0] + c[1] + c[2] + c[3] + c[4] + c[5] + c[6] + c[7];
        float tot = __shfl(s, 0, 32) + __shfl(s, 16, 32);
        if (lane == 0) out[0] = -tot;
    }
}

extern "C" void kernel_launch(void* const* d_in, const int* in_sizes, int n_in,
                              void* d_out, int out_size, void* d_ws, size_t ws_size,
                              hipStream_t stream) {
    const float* logprobs = (const float*)d_in[0];
    const int*   label    = (const int*)  d_in[1];
    const float* reward   = (const float*)d_in[2];
    // d_in[3] = use_cuda scalar, ignored.
    float* out = (float*)d_out;

    const int n     = in_sizes[1];            // bsz*seqlen = 8192
    const int vocab = in_sizes[0] / n;        // 32000

    pg_loss_kernel<<<1, 1024, 0, stream>>>(logprobs, label, reward, out, n, vocab);
}